// Gin_37366215475921
// MI455X (gfx1250) — compile-verified
//
#include <hip/hip_runtime.h>

#define N_NODES 50000
#define N_EDGES 800000
#define DIM     128
#define N_LAYERS 5
#define BN_EPS  1e-3f

#define KP_STRIDE  144   // v2f units per K-pair row of staged W (bank-conflict-free pad)
#define ZROW_STRIDE 132  // floats per row of staged Z tile (bank-conflict-free pad)

typedef __attribute__((ext_vector_type(2))) float v2f;
typedef __attribute__((ext_vector_type(8))) float v8f;

// pooled = (1 + eps[layer]) * h
__global__ void __launch_bounds__(256) gin_init_pooled(const float* __restrict__ h,
                                                       float* __restrict__ pooled,
                                                       const float* __restrict__ eps,
                                                       int layer) {
  int t = blockIdx.x * blockDim.x + threadIdx.x;
  const int total4 = N_NODES * DIM / 4;
  if (t < total4) {
    float s = 1.0f + eps[layer];
    float4 v = ((const float4*)h)[t];
    float4 o;
    o.x = s * v.x; o.y = s * v.y; o.z = s * v.z; o.w = s * v.w;
    ((float4*)pooled)[t] = o;
  }
}

// pooled[dst] += h[src]  (segment-sum via L2-resident float atomics)
__global__ void __launch_bounds__(256) gin_scatter(const float* __restrict__ h,
                                                   const int* __restrict__ esrc,
                                                   const int* __restrict__ edst,
                                                   float* __restrict__ pooled) {
  long long t = (long long)blockIdx.x * blockDim.x + threadIdx.x;
  int e = (int)(t >> 5);
  if (e >= N_EDGES) return;
  int c = ((int)t & 31) * 4;
  int s = esrc[e];
  int d = edst[e];
  float4 v = *(const float4*)(h + (long long)s * DIM + c);
  float* p = pooled + (long long)d * DIM + c;
  atomicAdd(p + 0, v.x);
  atomicAdd(p + 1, v.y);
  atomicAdd(p + 2, v.z);
  atomicAdd(p + 3, v.w);
}

// Fused MLP+BN+ReLU with fp32 WMMA 16x16x4.
// 128 threads = 4 waves; each wave owns 32 rows (two 16-row C tiles sharing B frags).
__global__ void __launch_bounds__(128) gin_mlp(const float* __restrict__ pooled,
                                               const float* __restrict__ W1,
                                               const float* __restrict__ b1,
                                               const float* __restrict__ W2,
                                               const float* __restrict__ b2,
                                               const float* __restrict__ gamma,
                                               const float* __restrict__ beta,
                                               const float* __restrict__ mean,
                                               const float* __restrict__ var,
                                               float* __restrict__ hout) {
  // Staged weights, K-pair interleaved: sW2[kp*KP_STRIDE + n] = {W[2kp][n], W[2kp+1][n]}
  __shared__ v2f   sW2[64 * KP_STRIDE];              // ~73.7 KB
  __shared__ float sZ[4 * 32 * ZROW_STRIDE];         // ~67.6 KB per-wave relu(z1) tiles

  const int tid  = threadIdx.x;
  const int lane = tid & 31;
  const int wave = tid >> 5;      // 0..3
  const int lo   = lane & 15;     // M (A/C) or N (B) index within tile
  const int hi   = lane >> 4;     // K-half select for A/B frags, +8 rows for C/D
  const int row0 = blockIdx.x * 128 + wave * 32;  // first row of subtile 0
  const int row1 = row0 + 16;                     // first row of subtile 1
  const bool valid0 = row0 < N_NODES;             // N_NODES % 16 == 0
  const bool valid1 = row1 < N_NODES;

  // Cooperative stage of a 128x128 weight matrix into the interleaved LDS layout.
  auto stageW = [&](const float* __restrict__ Wg) {
#pragma unroll
    for (int i = 0; i < 32; ++i) {
      int j  = tid + i * 128;        // 0..4095
      int kp = j >> 6;               // 0..63
      int n0 = (j & 63) * 2;         // 0,2,..,126
      float2 a = *(const float2*)(Wg + (2 * kp) * DIM + n0);
      float2 b = *(const float2*)(Wg + (2 * kp + 1) * DIM + n0);
      float4 w; w.x = a.x; w.y = b.x; w.z = a.y; w.w = b.y;
      *(float4*)(&sW2[kp * KP_STRIDE + n0]) = w;
    }
  };

  stageW(W1);
  __syncthreads();

  // ---- GEMM1: Z = relu(pooled @ W1 + b1) -> per-wave LDS tile ----
  if (valid0) {
    const int ra0 = row0 + lo;
    const int ra1 = valid1 ? (row1 + lo) : ra0;   // clamp OOB subtile reads
    const v2f* pr0 = (const v2f*)(pooled + (long long)ra0 * DIM);
    const v2f* pr1 = (const v2f*)(pooled + (long long)ra1 * DIM);
    float* zt = sZ + wave * (32 * ZROW_STRIDE);
    for (int nt = 0; nt < 8; ++nt) {
      const int n = nt * 16 + lo;
      const float bias = b1[n];
      v8f acc0 = {bias, bias, bias, bias, bias, bias, bias, bias};
      v8f acc1 = acc0;
#pragma unroll
      for (int kb = 0; kb < 32; ++kb) {
        const int kp = kb * 2 + hi;            // K-pair index: k = 2*kp
        v2f bf = sW2[kp * KP_STRIDE + n];      // {W[k][n], W[k+1][n]}
        v2f a0 = pr0[kp];                      // {P[m0][k], P[m0][k+1]}
        v2f a1 = pr1[kp];
        acc0 = __builtin_amdgcn_wmma_f32_16x16x4_f32(false, a0, false, bf,
                                                     (short)0, acc0, false, false);
        acc1 = __builtin_amdgcn_wmma_f32_16x16x4_f32(false, a1, false, bf,
                                                     (short)0, acc1, false, false);
      }
#pragma unroll
      for (int r = 0; r < 8; ++r) {
        zt[(r + 8 * hi) * ZROW_STRIDE + n]        = fmaxf(acc0[r], 0.0f);
        zt[(16 + r + 8 * hi) * ZROW_STRIDE + n]   = fmaxf(acc1[r], 0.0f);
      }
    }
  }
  __syncthreads();

  stageW(W2);
  __syncthreads();

  // ---- GEMM2: Z @ W2 + b2 -> BN -> relu -> global ----
  if (valid0) {
    const v2f* z0 = (const v2f*)(sZ + wave * (32 * ZROW_STRIDE) + lo * ZROW_STRIDE);
    const v2f* z1 = (const v2f*)(sZ + wave * (32 * ZROW_STRIDE) + (16 + lo) * ZROW_STRIDE);
    for (int nt = 0; nt < 8; ++nt) {
      const int n = nt * 16 + lo;
      const float bias = b2[n];
      v8f acc0 = {bias, bias, bias, bias, bias, bias, bias, bias};
      v8f acc1 = acc0;
#pragma unroll
      for (int kb = 0; kb < 32; ++kb) {
        const int kp = kb * 2 + hi;
        v2f bf = sW2[kp * KP_STRIDE + n];
        v2f a0 = z0[kp];
        v2f a1 = z1[kp];
        acc0 = __builtin_amdgcn_wmma_f32_16x16x4_f32(false, a0, false, bf,
                                                     (short)0, acc0, false, false);
        acc1 = __builtin_amdgcn_wmma_f32_16x16x4_f32(false, a1, false, bf,
                                                     (short)0, acc1, false, false);
      }
      const float gm  = gamma[n];
      const float bt  = beta[n];
      const float mu  = mean[n];
      const float inv = rsqrtf(var[n] + BN_EPS);
#pragma unroll
      for (int r = 0; r < 8; ++r) {
        float z = gm * (acc0[r] - mu) * inv + bt;
        hout[(long long)(row0 + r + 8 * hi) * DIM + n] = fmaxf(z, 0.0f);
      }
      if (valid1) {
#pragma unroll
        for (int r = 0; r < 8; ++r) {
          float z = gm * (acc1[r] - mu) * inv + bt;
          hout[(long long)(row1 + r + 8 * hi) * DIM + n] = fmaxf(z, 0.0f);
        }
      }
    }
  }
}

extern "C" void kernel_launch(void* const* d_in, const int* in_sizes, int n_in,
                              void* d_out, int out_size, void* d_ws, size_t ws_size,
                              hipStream_t stream) {
  const float* x     = (const float*)d_in[0];
  const int*   esrc  = (const int*)d_in[1];
  const int*   edst  = (const int*)d_in[2];
  const float* eps   = (const float*)d_in[3];
  const float* W1    = (const float*)d_in[4];
  const float* b1    = (const float*)d_in[5];
  const float* W2    = (const float*)d_in[6];
  const float* b2    = (const float*)d_in[7];
  const float* gamma = (const float*)d_in[8];
  const float* beta  = (const float*)d_in[9];
  const float* mean  = (const float*)d_in[10];
  const float* var   = (const float*)d_in[11];

  float* out    = (float*)d_out;
  float* pooled = (float*)d_ws;  // N_NODES*DIM floats = 25.6 MB scratch

  const size_t slot = (size_t)N_NODES * DIM;

  // hidden_rep[0] = x  (output slots double as the layer ping-pong buffers)
  hipMemcpyAsync(out, x, slot * sizeof(float), hipMemcpyDeviceToDevice, stream);

  const int initBlocks = (N_NODES * DIM / 4 + 255) / 256;
  const int scatBlocks = (int)(((long long)N_EDGES * 32 + 255) / 256);
  const int mlpBlocks  = (N_NODES + 127) / 128;

  for (int l = 0; l < N_LAYERS; ++l) {
    const float* h  = out + (size_t)l * slot;
    float*       hn = out + (size_t)(l + 1) * slot;

    gin_init_pooled<<<initBlocks, 256, 0, stream>>>(h, pooled, eps, l);
    gin_scatter<<<scatBlocks, 256, 0, stream>>>(h, esrc, edst, pooled);
    gin_mlp<<<mlpBlocks, 128, 0, stream>>>(
        pooled,
        W1 + (size_t)l * DIM * DIM, b1 + (size_t)l * DIM,
        W2 + (size_t)l * DIM * DIM, b2 + (size_t)l * DIM,
        gamma + (size_t)l * DIM, beta + (size_t)l * DIM,
        mean + (size_t)l * DIM, var + (size_t)l * DIM,
        hn);
  }
}